// SubdivideMeshes_10892037062803
// MI455X (gfx1250) — compile-verified
//
#include <hip/hip_runtime.h>
#include <hip/hip_bf16.h>
#include <stdint.h>

// ---------------------------------------------------------------------------
// SubdivideMeshes for MI455X (gfx1250).
//
// Pure bandwidth problem (~0.2 FLOP/byte => WMMA inapplicable): levers are
// NT stores (keep the 96MB verts working set resident in the 192MB L2 for the
// ~6x-reuse midpoint gather), the CDNA5 async global<->LDS DMA path for the
// streaming passthrough copy (b128 loads / b32 stores, dst is only 4B-aligned
// because (V+E)*3 is odd), and global_prefetch for the index streams.
//
// Inputs (setup_inputs order, JAX x64 off => indices are int32):
//   d_in[0] verts (N,V,3) f32   d_in[1] faces (F,3) i32
//   d_in[2] edges (E,2) i32     d_in[3] face_to_edge (F,3) i32
// Output (flattened, float): new_verts (N,V+E,3) then new_faces (4F,3).
// ---------------------------------------------------------------------------

// ---- Kernel 1: verts passthrough copy via async global->LDS->global DMA ----
// Each block stages 2048 floats (8KB) through LDS as a verbatim mirror of the
// global chunk. Loads: 2 x b128 per lane (16B-aligned on src). Stores: 8 x b32
// per lane (widest width legal on the odd-offset dst). All transfers are fully
// coalesced 128B+/wave requests tracked on ASYNCcnt; default (RT) load hint
// deliberately warms L2 with verts for the midpoint gather that follows.
__global__ void __launch_bounds__(256)
copy_verts_async(const float* __restrict__ src, float* __restrict__ dst,
                 long long vStride /*floats per batch, src*/,
                 long long oStride /*floats per batch, dst*/) {
    __shared__ float buf[2048];  // 8 KB staging, mirrors global chunk
    const int t = threadIdx.x;
    const long long n    = (long long)blockIdx.y;
    const long long base = (long long)blockIdx.x * 2048;

    const float* gsrc128 = src + n * vStride + base + (long long)t * 4;  // 16B-aligned
    float*       gdst    = dst + n * oStride + base + t;                 // 4B-aligned
    unsigned lds128 = (unsigned)(uintptr_t)(&buf[t * 4]);  // byte off t*16
    unsigned lds32  = (unsigned)(uintptr_t)(&buf[t]);      // byte off t*4

    // 2 async b128 DMA loads into LDS (lane l covers bytes l*16 + {0,4096})
    asm volatile(
        "global_load_async_to_lds_b128 %0, %1, off\n\t"
        "global_load_async_to_lds_b128 %0, %1, off offset:4096"
        :: "v"(lds128), "v"(gsrc128) : "memory");

    asm volatile("s_wait_asynccnt 0" ::: "memory");

    // 8 async b32 DMA stores from LDS back to global (NT: streamed output)
    asm volatile(
        "global_store_async_from_lds_b32 %0, %1, off th:TH_STORE_NT\n\t"
        "global_store_async_from_lds_b32 %0, %1, off offset:1024 th:TH_STORE_NT\n\t"
        "global_store_async_from_lds_b32 %0, %1, off offset:2048 th:TH_STORE_NT\n\t"
        "global_store_async_from_lds_b32 %0, %1, off offset:3072 th:TH_STORE_NT\n\t"
        "global_store_async_from_lds_b32 %0, %1, off offset:4096 th:TH_STORE_NT\n\t"
        "global_store_async_from_lds_b32 %0, %1, off offset:5120 th:TH_STORE_NT\n\t"
        "global_store_async_from_lds_b32 %0, %1, off offset:6144 th:TH_STORE_NT\n\t"
        "global_store_async_from_lds_b32 %0, %1, off offset:7168 th:TH_STORE_NT"
        :: "v"(gdst), "v"(lds32) : "memory");
    // S_ENDPGM performs an implicit wait-idle (ASYNCcnt drains before exit).
}

// Scalar tail for the (here empty) remainder of the copy.
__global__ void copy_verts_tail(const float* __restrict__ src, float* __restrict__ dst,
                                long long vStride, long long oStride,
                                long long start, long long count) {
    long long i = start + (long long)blockIdx.x * blockDim.x + threadIdx.x;
    if (i >= start + count) return;
    long long n = (long long)blockIdx.y;
    __builtin_nontemporal_store(src[n * vStride + i], &dst[n * oStride + i]);
}

// ---- Kernel 2: edge midpoints -------------------------------------------
// One thread per edge, batch loop inside: edge indices are loaded once (b64)
// and the 8 gathers per endpoint ride the L2-resident verts array (edges are
// sorted by v0, so gathers are near-sequential). NT stores keep ~300MB of
// output from evicting the gather working set.
__global__ void __launch_bounds__(256)
midpoint_kernel(const float* __restrict__ verts, const int2* __restrict__ edges,
                float* __restrict__ out, long long E,
                long long vStride, long long oStride, long long V, int N) {
    long long e = (long long)blockIdx.x * blockDim.x + threadIdx.x;
    if (e >= E) return;
    __builtin_prefetch(edges + e + 8192, 0, 0);  // -> global_prefetch_b8

    const int2 vp = edges[e];
    const long long a = 3LL * (long long)vp.x;
    const long long b = 3LL * (long long)vp.y;
    float* d0 = out + (V + e) * 3LL;

    for (int n = 0; n < N; ++n) {
        const float* s = verts + (long long)n * vStride;
        const float x = 0.5f * (s[a + 0] + s[b + 0]);
        const float y = 0.5f * (s[a + 1] + s[b + 1]);
        const float z = 0.5f * (s[a + 2] + s[b + 2]);
        float* d = d0 + (long long)n * oStride;
        __builtin_nontemporal_store(x, d + 0);
        __builtin_nontemporal_store(y, d + 1);
        __builtin_nontemporal_store(z, d + 2);
    }
}

// ---- Kernel 3: subdivided face topology ---------------------------------
// Indices < V+E (~4.19M) < 2^24 are exact in fp32, matching the float d_out.
__global__ void __launch_bounds__(256)
faces_kernel(const int* __restrict__ faces, const int* __restrict__ fte,
             float* __restrict__ out, long long F, float Vf) {
    long long f = (long long)blockIdx.x * blockDim.x + threadIdx.x;
    if (f >= F) return;
    __builtin_prefetch(faces + 3 * (f + 8192), 0, 0);
    __builtin_prefetch(fte   + 3 * (f + 8192), 0, 0);

    const float v0 = (float)faces[3 * f + 0];
    const float v1 = (float)faces[3 * f + 1];
    const float v2 = (float)faces[3 * f + 2];
    const float e0 = (float)fte[3 * f + 0] + Vf;
    const float e1 = (float)fte[3 * f + 1] + Vf;
    const float e2 = (float)fte[3 * f + 2] + Vf;

    float* o0 = out + 3 * f;                 // f0 = [v0, e2, e1]
    float* o1 = out + 3 * (F + f);           // f1 = [v1, e0, e2]
    float* o2 = out + 3 * (2 * F + f);       // f2 = [v2, e1, e0]
    float* o3 = out + 3 * (3 * F + f);       // f3 = [e0, e1, e2]
    __builtin_nontemporal_store(v0, o0 + 0);
    __builtin_nontemporal_store(e2, o0 + 1);
    __builtin_nontemporal_store(e1, o0 + 2);
    __builtin_nontemporal_store(v1, o1 + 0);
    __builtin_nontemporal_store(e0, o1 + 1);
    __builtin_nontemporal_store(e2, o1 + 2);
    __builtin_nontemporal_store(v2, o2 + 0);
    __builtin_nontemporal_store(e1, o2 + 1);
    __builtin_nontemporal_store(e0, o2 + 2);
    __builtin_nontemporal_store(e0, o3 + 0);
    __builtin_nontemporal_store(e1, o3 + 1);
    __builtin_nontemporal_store(e2, o3 + 2);
}

extern "C" void kernel_launch(void* const* d_in, const int* in_sizes, int n_in,
                              void* d_out, int out_size, void* d_ws, size_t ws_size,
                              hipStream_t stream) {
    const float* verts = (const float*)d_in[0];
    const int*   faces = (const int*)d_in[1];
    const int*   edges = (const int*)d_in[2];
    const int*   fte   = (const int*)d_in[3];
    float*       out   = (float*)d_out;

    const long long N = 8;  // BATCH constant in the reference
    const long long F = (long long)in_sizes[1] / 3;
    const long long E = (long long)in_sizes[2] / 2;
    const long long V = (long long)in_sizes[0] / (3 * N);
    const long long vStride = V * 3;            // floats per batch, input
    const long long oStride = (V + E) * 3;      // floats per batch, new_verts

    // 1) passthrough copy of verts, async-DMA staged through LDS
    const long long perBatch   = V * 3;
    const long long blkFloats  = 2048;
    const long long fullBlocks = perBatch / blkFloats;   // 1536 here (exact)
    if (fullBlocks > 0) {
        dim3 g((unsigned)fullBlocks, (unsigned)N);
        copy_verts_async<<<g, 256, 0, stream>>>(verts, out, vStride, oStride);
    }
    const long long rem = perBatch - fullBlocks * blkFloats;
    if (rem > 0) {
        dim3 g((unsigned)((rem + 255) / 256), (unsigned)N);
        copy_verts_tail<<<g, 256, 0, stream>>>(verts, out, vStride, oStride,
                                               fullBlocks * blkFloats, rem);
    }

    // 2) edge midpoints -> new_verts[:, V:, :]
    {
        dim3 g((unsigned)((E + 255) / 256));
        midpoint_kernel<<<g, 256, 0, stream>>>(verts, (const int2*)edges, out,
                                               E, vStride, oStride, V, (int)N);
    }

    // 3) new faces, placed after all new_verts in d_out
    {
        float* outFaces = out + N * oStride;
        dim3 g((unsigned)((F + 255) / 256));
        faces_kernel<<<g, 256, 0, stream>>>(faces, fte, outFaces, F, (float)V);
    }
}